// GromovWasserstein_56573309223311
// MI455X (gfx1250) — compile-verified
//
#include <hip/hip_runtime.h>

typedef __attribute__((ext_vector_type(2))) float v2f;
typedef __attribute__((ext_vector_type(8))) float v8f;

#define GW_N 512
#define GW_D 128

// ---------------------------------------------------------------------------
// Kernel 1: squared norms of every row of source (rows 0..511) and target
// (rows 512..1023). One wave32 per row.
// ---------------------------------------------------------------------------
__global__ void gw_norms(const float* __restrict__ src,
                         const float* __restrict__ tgt,
                         float* __restrict__ ns) {
  const int wave = (blockIdx.x * blockDim.x + threadIdx.x) >> 5;
  const int lane = threadIdx.x & 31;
  if (wave >= 2 * GW_N) return;
  const float* X = (wave < GW_N) ? src : tgt;
  const float* p = X + (size_t)(wave & (GW_N - 1)) * GW_D;
  float s = 0.f;
#pragma unroll
  for (int q = 0; q < GW_D; q += 32) {
    float x = p[q + lane];
    s += x * x;
  }
#pragma unroll
  for (int m = 16; m >= 1; m >>= 1) s += __shfl_xor(s, m, 32);
  if (lane == 0) ns[wave] = s;
}

// ---------------------------------------------------------------------------
// Kernel 2: per 16-row tile of one distance matrix (source or target):
//   G = X * X^T via V_WMMA_F32_16X16X4_F32,  C = ns_i + ns_j - 2G
// Accumulate rowsum(C) and sum(C^2) for the owned rows. 64 blocks x 1 wave.
// ---------------------------------------------------------------------------
__global__ void gw_tiles(const float* __restrict__ src,
                         const float* __restrict__ tgt,
                         const float* __restrict__ ns,
                         float* __restrict__ rs,
                         float* __restrict__ psq) {
  const int b   = blockIdx.x;   // 0..63
  const int mat = b >> 5;       // 0 = source, 1 = target
  const int rb  = b & 31;       // which 16-row tile
  const float* X   = mat ? tgt : src;
  const float* nsb = ns + mat * GW_N;

  const int lane = threadIdx.x & 31;
  const int lo   = lane & 15;   // M (for A) / N (for B/C/D) within tile
  const int hi   = lane >> 4;   // which K-half (A/B) / which M-half (C/D)
  const int r0   = rb * 16;

  // Preload the A tile rows (row r0+lo) for all K in WMMA f32 16x16x4 layout:
  // per k-step kk, lane holds K = 4*kk + 2*hi + {0,1}.
  v2f a[32];
  const float* pa = X + (size_t)(r0 + lo) * GW_D + 2 * hi;
#pragma unroll
  for (int kk = 0; kk < 32; ++kk) a[kk] = *(const v2f*)(pa + 4 * kk);

  // ns for the rows this lane's C/D slots map to (VGPR s -> M = s + 8*hi).
  float nsr[8];
#pragma unroll
  for (int s = 0; s < 8; ++s) nsr[s] = nsb[r0 + s + 8 * hi];

  float rsacc[8];
#pragma unroll
  for (int s = 0; s < 8; ++s) rsacc[s] = 0.f;
  float ss = 0.f;

  for (int jt = 0; jt < 32; ++jt) {
    const int c0 = jt * 16;
    // B tile: B[k][n] = X[c0+n][k], mirrored layout of A with N across lanes.
    const float* pb = X + (size_t)(c0 + lo) * GW_D + 2 * hi;
    const float nsc = nsb[c0 + lo];

    v8f acc = {};
#pragma unroll
    for (int kk = 0; kk < 32; ++kk) {
      v2f bv = *(const v2f*)(pb + 4 * kk);
      acc = __builtin_amdgcn_wmma_f32_16x16x4_f32(
          /*neg_a=*/false, a[kk], /*neg_b=*/false, bv,
          /*c_mod=*/(short)0, acc, /*reuse_a=*/false, /*reuse_b=*/false);
    }

#pragma unroll
    for (int s = 0; s < 8; ++s) {
      float dv = nsr[s] + nsc - 2.f * acc[s];
      rsacc[s] += dv;
      ss += dv * dv;
    }
  }

  // Row sums: reduce across the 16 lanes sharing each M; write once per row.
#pragma unroll
  for (int s = 0; s < 8; ++s) {
    float v = rsacc[s];
    v += __shfl_xor(v, 1, 32);
    v += __shfl_xor(v, 2, 32);
    v += __shfl_xor(v, 4, 32);
    v += __shfl_xor(v, 8, 32);
    if (lo == 0) rs[mat * GW_N + r0 + s + 8 * hi] = v;
  }

  // Sum of squares: reduce across the whole wave; one partial per block.
#pragma unroll
  for (int m = 16; m >= 1; m >>= 1) ss += __shfl_xor(ss, m, 32);
  if (lane == 0) psq[b] = ss;
}

// ---------------------------------------------------------------------------
// Kernel 3: gw = ( n*(sum C_s^2 + sum C_t^2) - 2*sum_i rs_s[i]*rs_t[i] ) / n^2
// ---------------------------------------------------------------------------
__global__ void gw_final(const float* __restrict__ rs,
                         const float* __restrict__ psq,
                         float* __restrict__ out) {
  __shared__ float sm[GW_N];
  const int t = threadIdx.x;
  sm[t] = rs[t] * rs[t + GW_N];
  __syncthreads();
  for (int off = GW_N / 2; off >= 1; off >>= 1) {
    if (t < off) sm[t] += sm[t + off];
    __syncthreads();
  }
  if (t == 0) {
    float ssum = 0.f;
    for (int j = 0; j < 64; ++j) ssum += psq[j];
    const float n = (float)GW_N;
    out[0] = (n * ssum - 2.f * sm[0]) / (n * n);
  }
}

// ---------------------------------------------------------------------------
extern "C" void kernel_launch(void* const* d_in, const int* in_sizes, int n_in,
                              void* d_out, int out_size, void* d_ws, size_t ws_size,
                              hipStream_t stream) {
  (void)in_sizes; (void)n_in; (void)out_size; (void)ws_size;
  const float* src = (const float*)d_in[0];
  const float* tgt = (const float*)d_in[1];
  float* ws  = (float*)d_ws;
  float* ns  = ws;          // 1024 floats: row squared-norms (src, tgt)
  float* rsv = ws + 1024;   // 1024 floats: rowsums of C_s, C_t
  float* psq = ws + 2048;   //   64 floats: per-tile-block sum(C^2) partials

  gw_norms<<<(2 * GW_N) / 8, 256, 0, stream>>>(src, tgt, ns);
  gw_tiles<<<64, 32, 0, stream>>>(src, tgt, ns, rsv, psq);
  gw_final<<<1, GW_N, 0, stream>>>(rsv, psq, (float*)d_out);
}